// AutoInt_51651276702509
// MI455X (gfx1250) — compile-verified
//
#include <hip/hip_runtime.h>
#include <hip/hip_bf16.h>

// Problem dimensions (fixed by the reference)
#define BATCH 4096
#define FDIM  128
#define LNUM  3
#define H1DIM 256
#define H2DIM 128

typedef __attribute__((ext_vector_type(2))) float v2f;
typedef __attribute__((ext_vector_type(8))) float v8f;

// ---------------------------------------------------------------------------
// Kernel 1: 3 stacked InteractingLayers with scalar (1x1) projections.
// One block per batch row, 128 threads (thread t owns field f=t).
//   out_f = wv * sum_g softmax_g(wq*wk*x_f*x_g) * x_g + wr * x_f ; relu
// Softmax stabilized per-thread: max_g(sf*x_g) = sf>=0 ? sf*xmax : sf*xmin.
// ---------------------------------------------------------------------------
__global__ __launch_bounds__(FDIM)
void attn_layers_kernel(const float* __restrict__ X,
                        const float* __restrict__ wq,
                        const float* __restrict__ wk,
                        const float* __restrict__ wv,
                        const float* __restrict__ wr,
                        float* __restrict__ H)
{
    __shared__ float xs[FDIM];
    __shared__ float rmx[FDIM];
    __shared__ float rmn[FDIM];

    const int b = blockIdx.x;
    const int t = threadIdx.x;

    float x = X[b * FDIM + t];

    #pragma unroll 1
    for (int l = 0; l < LNUM; ++l) {
        const float cqk = wq[l] * wk[l];
        const float wvl = wv[l];
        const float wrl = wr[l];

        xs[t]  = x;
        rmx[t] = x;
        rmn[t] = x;
        __syncthreads();

        // block-wide max/min of x
        #pragma unroll
        for (int s = FDIM / 2; s > 0; s >>= 1) {
            if (t < s) {
                rmx[t] = fmaxf(rmx[t], rmx[t + s]);
                rmn[t] = fminf(rmn[t], rmn[t + s]);
            }
            __syncthreads();
        }
        const float xmax = rmx[0];
        const float xmin = rmn[0];

        const float sf = cqk * x;                      // q_f * wk (folded)
        const float m  = (sf >= 0.0f) ? sf * xmax : sf * xmin;

        float denom = 0.0f, numer = 0.0f;
        #pragma unroll 4
        for (int g = 0; g < FDIM; ++g) {               // LDS broadcast reads
            const float xg = xs[g];
            const float e  = __expf(sf * xg - m);
            denom += e;
            numer += e * xg;
        }

        const float out = wvl * (numer / denom) + wrl * x;
        x = fmaxf(out, 0.0f);
        __syncthreads();                               // done reading xs before next layer rewrites
    }

    H[b * FDIM + t] = x;
}

// ---------------------------------------------------------------------------
// Kernel 2/3: C[M,N] = relu(A[M,K] @ W[K,N] + bias) using V_WMMA_F32_16X16X4_F32.
// Block = 128 threads (4 waves). Block tile: 16(M) x 64(N); each wave owns a
// 16x16 tile and issues K/4 WMMAs. A-slab (16 x K) staged in LDS with padded
// stride to avoid bank conflicts. Grid: (M/16, N/64).
// ---------------------------------------------------------------------------
template<int KDIM, int NDIM, bool RELU>
__global__ __launch_bounds__(128)
void gemm_bias_relu_wmma(const float* __restrict__ A,
                         const float* __restrict__ W,
                         const float* __restrict__ bias,
                         float* __restrict__ C)
{
    constexpr int KP = KDIM + 4;                       // padded LDS stride
    __shared__ float As[16 * KP];

    const int m0 = blockIdx.x * 16;
    const int wave = threadIdx.x >> 5;
    const int n0 = blockIdx.y * 64 + wave * 16;

    // Cooperative stage of the 16 x KDIM A-slab (fully coalesced).
    #pragma unroll
    for (int r = 0; r < 16; ++r)
        for (int c = threadIdx.x; c < KDIM; c += 128)
            As[r * KP + c] = A[(m0 + r) * KDIM + c];
    __syncthreads();

    const int lane = threadIdx.x & 31;
    const int ln   = lane & 15;                        // M (for A) / N (for B) index
    const int kh   = (lane >> 4) * 2;                  // K sub-offset from lane half

    v8f acc = {};
    #pragma unroll 8
    for (int k = 0; k < KDIM; k += 4) {
        v2f a, bfrag;
        a.x     = As[ln * KP + k + kh];
        a.y     = As[ln * KP + k + kh + 1];
        bfrag.x = W[(k + kh) * NDIM + n0 + ln];
        bfrag.y = W[(k + kh + 1) * NDIM + n0 + ln];
        // D = A(16x4) * B(4x16) + C   (fp32, exact vs reference)
        acc = __builtin_amdgcn_wmma_f32_16x16x4_f32(
            /*neg_a=*/false, a, /*neg_b=*/false, bfrag,
            /*c_mod=*/(short)0, acc, /*reuse_a=*/false, /*reuse_b=*/false);
    }

    const float bv = bias[n0 + ln];
    const int rbase = m0 + (lane >> 4) * 8;
    #pragma unroll
    for (int r = 0; r < 8; ++r) {
        float v = acc[r] + bv;
        if (RELU) v = fmaxf(v, 0.0f);
        C[(rbase + r) * NDIM + n0 + ln] = v;
    }
}

// ---------------------------------------------------------------------------
// Kernel 4: out[b] = dot(H2[b,:], Wf)   (H2DIM = 128)
// ---------------------------------------------------------------------------
__global__ __launch_bounds__(H2DIM)
void final_dot_kernel(const float* __restrict__ H2,
                      const float* __restrict__ Wf,
                      float* __restrict__ out)
{
    __shared__ float red[H2DIM];
    const int b = blockIdx.x;
    const int t = threadIdx.x;

    red[t] = H2[b * H2DIM + t] * Wf[t];
    __syncthreads();
    #pragma unroll
    for (int s = H2DIM / 2; s > 0; s >>= 1) {
        if (t < s) red[t] += red[t + s];
        __syncthreads();
    }
    if (t == 0) out[b] = red[0];
}

// ---------------------------------------------------------------------------
// Launch: X, wq, wk, wv, wr, W1, b1, W2, b2, Wf  (setup_inputs order)
// Workspace layout: h0 (B*F) | h1 (B*H1) | h2 (B*H2)   = 8 MB of fp32
// ---------------------------------------------------------------------------
extern "C" void kernel_launch(void* const* d_in, const int* in_sizes, int n_in,
                              void* d_out, int out_size, void* d_ws, size_t ws_size,
                              hipStream_t stream)
{
    const float* X  = (const float*)d_in[0];
    const float* wq = (const float*)d_in[1];
    const float* wk = (const float*)d_in[2];
    const float* wv = (const float*)d_in[3];
    const float* wr = (const float*)d_in[4];
    const float* W1 = (const float*)d_in[5];
    const float* b1 = (const float*)d_in[6];
    const float* W2 = (const float*)d_in[7];
    const float* b2 = (const float*)d_in[8];
    const float* Wf = (const float*)d_in[9];
    float* out = (float*)d_out;

    float* h0 = (float*)d_ws;                 // (B, F)   attention output
    float* h1 = h0 + (size_t)BATCH * FDIM;    // (B, H1)
    float* h2 = h1 + (size_t)BATCH * H1DIM;   // (B, H2)

    attn_layers_kernel<<<BATCH, FDIM, 0, stream>>>(X, wq, wk, wv, wr, h0);

    gemm_bias_relu_wmma<FDIM, H1DIM, true>
        <<<dim3(BATCH / 16, H1DIM / 64), 128, 0, stream>>>(h0, W1, b1, h1);

    gemm_bias_relu_wmma<H1DIM, H2DIM, true>
        <<<dim3(BATCH / 16, H2DIM / 64), 128, 0, stream>>>(h1, W2, b2, h2);

    final_dot_kernel<<<BATCH, H2DIM, 0, stream>>>(h2, Wf, out);
}